// FastDWT_28905129902549
// MI455X (gfx1250) — compile-verified
//
#include <hip/hip_runtime.h>
#include <hip/hip_bf16.h>
#include <cstdint>

// ---------------------------------------------------------------------------
// Haar 2x2 DWT, (4,64,512,512) f32 -> 4x (4,64,256,256) f32 (ll,lh,hl,hh).
// Pure streaming kernel (512 MB total traffic, ~0.4 flop/byte): bandwidth
// bound on MI455X (23.3 TB/s -> ~22us floor). Data path: CDNA5 async
// global->LDS DMA (GLOBAL_LOAD_ASYNC_TO_LDS_B128) with double buffering +
// s_wait_asynccnt + workgroup barriers; compute reads LDS with ds_load_b128;
// outputs use non-temporal 64-bit stores (zero reuse -> TH=NT).
// ---------------------------------------------------------------------------

typedef float v4f __attribute__((ext_vector_type(4)));
typedef float v2f __attribute__((ext_vector_type(2)));
// Must match the builtin's canonical param type: gcc-style vector_size(16) int.
typedef int v4i_vs __attribute__((vector_size(4 * sizeof(int))));

#define W_IN 512
#define HW_IN (512 * 512)
#define W_OUT 256
#define HW_OUT (256 * 256)
#define PLANES 256                 // B*C = 4*64
#define TILE_FLOATS 2048           // 4 input rows = 8 KB, contiguous in memory
#define TILES_PER_PLANE 128        // 512 rows / 4
#define TILES_TOTAL (PLANES * TILES_PER_PLANE)   // 32768
#define TILES_PER_BLOCK 8
#define NBLOCKS (TILES_TOTAL / TILES_PER_BLOCK)  // 4096
#define NSUB ((size_t)PLANES * HW_OUT)           // 16,777,216 elems per subband

#if defined(__AMDGCN__)
#define USE_ASYNC 1

#if __has_builtin(__builtin_amdgcn_global_load_async_to_lds_b128)
// Signature (probe-confirmed via diagnostic):
//   void (v4i AS1* gsrc, v4i AS3* ldsdst, imm int offset, imm int cpol)
#define ASYNC_LOAD_B128(lptr, gptr)                                            \
  __builtin_amdgcn_global_load_async_to_lds_b128(                              \
      (__attribute__((address_space(1))) v4i_vs*)(uintptr_t)(gptr),            \
      (__attribute__((address_space(3))) v4i_vs*)(uint32_t)(uintptr_t)(lptr),  \
      0, 0)
#else
// inline-asm fallback: VDST = LDS byte address VGPR, VADDR = 64-bit global
#define ASYNC_LOAD_B128(lptr, gptr)                                            \
  asm volatile("global_load_async_to_lds_b128 %0, %1, off" ::                  \
                   "v"((uint32_t)(uintptr_t)(lptr)),                           \
                   "v"((uint64_t)(uintptr_t)(gptr))                            \
               : "memory")
#endif

#if __has_builtin(__builtin_amdgcn_s_wait_asynccnt)
#define WAIT_ASYNC(n) __builtin_amdgcn_s_wait_asynccnt(n)
#else
#define WAIT_ASYNC(n) asm volatile("s_wait_asynccnt %0" ::"n"(n) : "memory")
#endif

#else  // host pass: body is never codegen'd, keep it compilable
#define USE_ASYNC 0
#define WAIT_ASYNC(n)
#endif

__global__ __launch_bounds__(256) void FastDWT_haar_tiled(
    const float* __restrict__ x, float* __restrict__ out) {
  const int t  = threadIdx.x;
  const int rp = t >> 7;   // row-pair within tile (0/1)
  const int c4 = t & 127;  // 4-input-column group within row

#if USE_ASYNC
  __shared__ __align__(16) float lds[2 * TILE_FLOATS];  // 16 KB, double-buffered

  // Prologue: stage tile 0 into buffer 0 (each lane: 2 async B128 = 32 B)
  {
    const int tile0 = blockIdx.x * TILES_PER_BLOCK;
    const float* src =
        x + (size_t)(tile0 >> 7) * HW_IN + (size_t)(tile0 & 127) * TILE_FLOATS;
    ASYNC_LOAD_B128(lds + t * 4, src + t * 4);
    ASYNC_LOAD_B128(lds + 1024 + t * 4, src + 1024 + t * 4);
  }
#endif

  for (int it = 0; it < TILES_PER_BLOCK; ++it) {
    const int tile  = blockIdx.x * TILES_PER_BLOCK + it;
    const int plane = tile >> 7;
    const int tp    = tile & 127;

    v4f top, bot;
#if USE_ASYNC
    float* buf = lds + (it & 1) * TILE_FLOATS;
    if (it + 1 < TILES_PER_BLOCK) {
      // Issue next tile into the other buffer (safe: that buffer's readers
      // all passed the trailing barrier of iteration it-1).
      const int ntile = tile + 1;
      const float* nsrc = x + (size_t)(ntile >> 7) * HW_IN +
                          (size_t)(ntile & 127) * TILE_FLOATS;
      float* nbuf = lds + ((it + 1) & 1) * TILE_FLOATS;
      ASYNC_LOAD_B128(nbuf + t * 4, nsrc + t * 4);
      ASYNC_LOAD_B128(nbuf + 1024 + t * 4, nsrc + 1024 + t * 4);
      WAIT_ASYNC(2);  // async loads retire in order -> tile `it` complete
    } else {
      WAIT_ASYNC(0);
    }
    __syncthreads();  // all waves' DMA for tile `it` visible in LDS

    top = *(const v4f*)(buf + rp * 1024 + c4 * 4);        // ds_load_b128
    bot = *(const v4f*)(buf + rp * 1024 + 512 + c4 * 4);  // ds_load_b128
#else
    const float* src = x + (size_t)plane * HW_IN + (size_t)tp * TILE_FLOATS;
    top = __builtin_nontemporal_load((const v4f*)(src + rp * 1024 + c4 * 4));
    bot = __builtin_nontemporal_load((const v4f*)(src + rp * 1024 + 512 + c4 * 4));
#endif

    // Haar butterfly for two 2x2 blocks: a=top.even b=top.odd c=bot.even d=bot.odd
    const float s0a = top.x + top.y, d0a = top.x - top.y;
    const float s1a = bot.x + bot.y, d1a = bot.x - bot.y;
    const float s0b = top.z + top.w, d0b = top.z - top.w;
    const float s1b = bot.z + bot.w, d1b = bot.z - bot.w;

    const v2f ll = {0.5f * (s0a + s1a), 0.5f * (s0b + s1b)};
    const v2f lh = {0.5f * (s0a - s1a), 0.5f * (s0b - s1b)};
    const v2f hl = {0.5f * (d0a + d1a), 0.5f * (d0b + d1b)};
    const v2f hh = {0.5f * (d0a - d1a), 0.5f * (d0b - d1b)};

    const size_t obase = (size_t)plane * HW_OUT +
                         (size_t)(tp * 2 + rp) * W_OUT + (size_t)c4 * 2;
    __builtin_nontemporal_store(ll, (v2f*)(out + obase));
    __builtin_nontemporal_store(lh, (v2f*)(out + NSUB + obase));
    __builtin_nontemporal_store(hl, (v2f*)(out + 2 * NSUB + obase));
    __builtin_nontemporal_store(hh, (v2f*)(out + 3 * NSUB + obase));

#if USE_ASYNC
    __syncthreads();  // all reads of buf done before it is overwritten
#endif
  }
}

extern "C" void kernel_launch(void* const* d_in, const int* in_sizes, int n_in,
                              void* d_out, int out_size, void* d_ws,
                              size_t ws_size, hipStream_t stream) {
  (void)in_sizes; (void)n_in; (void)out_size; (void)d_ws; (void)ws_size;
  const float* x = (const float*)d_in[0];
  float* out = (float*)d_out;
  FastDWT_haar_tiled<<<NBLOCKS, 256, 0, stream>>>(x, out);
}